// attention_Layers_81836306858151
// MI455X (gfx1250) — compile-verified
//
#include <hip/hip_runtime.h>
#include <hip/hip_bf16.h>

// RGCN 2-layer forward for MI455X (gfx1250).
// Dense transforms via V_WMMA_F32_16X16X4_F32 (exact fp32 matrix op, wave32),
// 1x4 register-blocked, weights staged in LDS (bank-pad 72).
// Sparse aggregation via pre-inverted per-(rel,dst) degree scaling + single
// fp32 atomic accumulator per layer.

typedef float v2f __attribute__((ext_vector_type(2)));
typedef float v8f __attribute__((ext_vector_type(8)));

#define NREL 16

// ---------------------------------------------------------------- utilities
__global__ void zero_f32(float* __restrict__ p, int n) {
  int i = blockIdx.x * blockDim.x + threadIdx.x;
  if (i < n) p[i] = 0.0f;
}

__global__ void relu_f32(float* __restrict__ p, int n) {
  int i = blockIdx.x * blockDim.x + threadIdx.x;
  if (i < n) p[i] = fmaxf(p[i], 0.0f);
}

__global__ void sigmoid_f32(float* __restrict__ p, int n) {
  int i = blockIdx.x * blockDim.x + threadIdx.x;
  if (i < n) p[i] = 1.0f / (1.0f + __expf(-p[i]));
}

// per-(relation,dst) in-degree, then in-place inversion 1/max(c,1)
__global__ void count_edges(const int* __restrict__ dst, const int* __restrict__ typ,
                            float* __restrict__ cnt, int E, int N) {
  int e = blockIdx.x * blockDim.x + threadIdx.x;
  if (e < E) atomicAdd(&cnt[typ[e] * N + dst[e]], 1.0f);
}

__global__ void invert_cnt(float* __restrict__ p, int n) {
  int i = blockIdx.x * blockDim.x + threadIdx.x;
  if (i < n) p[i] = 1.0f / fmaxf(p[i], 1.0f);
}

// ---------------------------------------------------------- WMMA f32 GEMM
// C[nrows x (NT*16)] = X[nrows x 64] @ W[64 x (NT*16)] (+ bias if non-null).
// One wave computes a full 16-row x NT*16-col strip: A fragment loaded once
// per K-step and reused across NT WMMAs (16*NT wmma per wave).
// W is staged in LDS with row stride 72 floats: lane-halves read rows k and
// k+2, bank sets {col} vs {col+16} -> conflict-free ds_load_b32.
template <int NT>
__global__ void gemm_wmma_f32(const float* __restrict__ X,
                              const float* __restrict__ W,
                              const float* __restrict__ bias,
                              float* __restrict__ C, int nrows) {
  constexpr int DOUT = NT * 16;
  constexpr int S = 72;  // padded LDS row stride (floats)
  __shared__ float lw[64 * S];

  // cooperative stage of W (64 x DOUT) into LDS
  for (int i = threadIdx.x; i < 64 * DOUT; i += blockDim.x) {
    const int kr = i / DOUT;
    const int kc = i - kr * DOUT;
    lw[kr * S + kc] = W[i];
  }
  __syncthreads();

  const int lane = threadIdx.x & 31;
  const int mt = (blockIdx.x * blockDim.x + threadIdx.x) >> 5;  // m-strip id
  const int mtiles = (nrows + 15) >> 4;
  if (mt >= mtiles) return;  // wave-uniform: EXEC all-ones for WMMA

  const int m0 = mt << 4;
  const int l15 = lane & 15;
  const int khalf = (lane >> 4) << 1;  // 0 or 2

  int arow = m0 + l15;
  if (arow >= nrows) arow = nrows - 1;  // clamped rows never stored

  v8f c[NT];
#pragma unroll
  for (int nt = 0; nt < NT; ++nt) {
    const float binit = bias ? bias[nt * 16 + l15] : 0.0f;
#pragma unroll
    for (int i = 0; i < 8; ++i) c[nt][i] = binit;
  }

  const float* __restrict__ ap = X + (size_t)arow * 64 + khalf;
  const float* lwp = lw + khalf * S + l15;  // rows {khalf, khalf+1} base

#pragma unroll
  for (int k0 = 0; k0 < 64; k0 += 4) {
    // A 16x4: lanes 0-15 -> K={k0,k0+1}, lanes 16-31 -> K={k0+2,k0+3}
    const v2f a = *(const v2f*)(ap + k0);
#pragma unroll
    for (int nt = 0; nt < NT; ++nt) {
      // B 4x16: VGPR0 = rows {k0, k0+2}, VGPR1 = rows {k0+1, k0+3}
      v2f b;
      b.x = lwp[k0 * S + nt * 16];
      b.y = lwp[k0 * S + S + nt * 16];
      c[nt] = __builtin_amdgcn_wmma_f32_16x16x4_f32(
          /*neg_a=*/false, a, /*neg_b=*/false, b,
          /*c_mod=*/(short)0, c[nt], /*reuse_a=*/false, /*reuse_b=*/false);
    }
  }

  // C layout: VGPR i -> row m0 + (lane>>4)*8 + i, col = nt*16 + (lane&15)
  const int crow = m0 + ((lane >> 4) << 3);
  float* __restrict__ cp = C + (size_t)crow * DOUT + l15;
#pragma unroll
  for (int nt = 0; nt < NT; ++nt) {
#pragma unroll
    for (int i = 0; i < 8; ++i) {
      if (crow + i < nrows) cp[(size_t)i * DOUT + nt * 16] = c[nt][i];
    }
  }
}

// ------------------------------------------------------- edge scatter (per r)
// out[dst] += xr[src] * inv[r*N+dst] for edges with type==r.
// Dim-parallel: 2^LOGD consecutive threads handle one edge -> coalesced.
template <int LOGD>
__global__ void scatter_rel(const int* __restrict__ src, const int* __restrict__ dst,
                            const int* __restrict__ typ, const float* __restrict__ xr,
                            const float* __restrict__ inv, float* __restrict__ out,
                            int E, int N, int r) {
  const unsigned t = blockIdx.x * blockDim.x + threadIdx.x;
  const unsigned e = t >> LOGD;
  const unsigned d = t & ((1u << LOGD) - 1u);
  if (e >= (unsigned)E) return;
  __builtin_prefetch(typ + e + 8192, 0, 0);  // global_prefetch_b8 (speculative)
  if (typ[e] != r) return;
  const int s = src[e];
  const int dn = dst[e];
  const float w = inv[r * N + dn];
  const float v = xr[((size_t)s << LOGD) + d] * w;
  atomicAdd(&out[((size_t)dn << LOGD) + d], v);
}

// ------------------------------------------------------------------- launch
static inline int cdiv(int a, int b) { return (a + b - 1) / b; }

extern "C" void kernel_launch(void* const* d_in, const int* in_sizes, int n_in,
                              void* d_out, int out_size, void* d_ws, size_t ws_size,
                              hipStream_t stream) {
  const int* edge_index = (const int*)d_in[0];
  const int E = in_sizes[0] / 2;
  const int* e_src = edge_index;
  const int* e_dst = edge_index + E;
  const int* e_typ = (const int*)d_in[1];
  const float* emb  = (const float*)d_in[2];
  const float* w1   = (const float*)d_in[3];   // [16,64,64]
  const float* r1   = (const float*)d_in[4];   // [64,64]
  const float* b1   = (const float*)d_in[5];   // [64]
  const float* w2   = (const float*)d_in[6];   // [16,64,16]
  const float* r2   = (const float*)d_in[7];   // [64,16]
  const float* b2   = (const float*)d_in[8];   // [16]
  float* out = (float*)d_out;                  // [N,16]

  const int DIN = 64, HID = 64, DL = 16;
  const int N = in_sizes[2] / DIN;             // 100000

  // workspace layout (floats)
  float* inv = (float*)d_ws;                   // NREL*N  (6.4 MB)
  float* xr  = inv + (size_t)NREL * N;         // N*64    (25.6 MB)
  float* h1  = xr + (size_t)N * HID;           // N*64    (25.6 MB)

  const int mtiles = cdiv(N, 16);
  const int gblocks = cdiv(mtiles, 8);         // 8 waves (m-strips) per block

  // ---- per-(relation,dst) inverse degrees (shared by both layers) ----
  zero_f32<<<cdiv(NREL * N, 256), 256, 0, stream>>>(inv, NREL * N);
  count_edges<<<cdiv(E, 256), 256, 0, stream>>>(e_dst, e_typ, inv, E, N);
  invert_cnt<<<cdiv(NREL * N, 256), 256, 0, stream>>>(inv, NREL * N);

  // ---- layer 1: h1 = emb@root1 + bias1 + sum_r mean_r(emb@W1[r]) ----
  gemm_wmma_f32<4><<<gblocks, 256, 0, stream>>>(emb, r1, b1, h1, N);
  for (int r = 0; r < NREL; ++r) {
    gemm_wmma_f32<4><<<gblocks, 256, 0, stream>>>(
        emb, w1 + (size_t)r * DIN * HID, nullptr, xr, N);
    scatter_rel<6><<<cdiv(E * 64, 256), 256, 0, stream>>>(
        e_src, e_dst, e_typ, xr, inv, h1, E, N, r);
  }
  relu_f32<<<cdiv(N * HID, 256), 256, 0, stream>>>(h1, N * HID);

  // ---- layer 2: out = h1@root2 + bias2 + sum_r mean_r(h1@W2[r]) ----
  gemm_wmma_f32<1><<<gblocks, 256, 0, stream>>>(h1, r2, b2, out, N);
  for (int r = 0; r < NREL; ++r) {
    gemm_wmma_f32<1><<<gblocks, 256, 0, stream>>>(
        h1, w2 + (size_t)r * HID * DL, nullptr, xr, N);
    scatter_rel<4><<<cdiv(E * 16, 256), 256, 0, stream>>>(
        e_src, e_dst, e_typ, xr, inv, out, E, N, r);
  }
  sigmoid_f32<<<cdiv(N * DL, 256), 256, 0, stream>>>(out, N * DL);
}